// RGCN_13846974562748
// MI455X (gfx1250) — compile-verified
//
#include <hip/hip_runtime.h>
#include <hip/hip_bf16.h>

#define N_NODES 50000
#define R_REL   4
#define E_EDGES 150000
#define DIM     128

typedef float v2f __attribute__((ext_vector_type(2)));
typedef float v8f __attribute__((ext_vector_type(8)));

// ---------------- utility: zero fill (grid-stride) ----------------
__global__ void rgcn_zero_f32(float* __restrict__ p, long n) {
    long i = (long)blockIdx.x * blockDim.x + threadIdx.x;
    long stride = (long)gridDim.x * blockDim.x;
    for (; i < n; i += stride) p[i] = 0.0f;
}

// ---------------- degree count: one thread per (relation, edge) ----------------
__global__ void rgcn_degree_count(const int* __restrict__ src,
                                  const int* __restrict__ dst,
                                  float* __restrict__ s_out,
                                  float* __restrict__ s_in) {
    int i = blockIdx.x * blockDim.x + threadIdx.x;  // i = r*E + e
    if (i >= R_REL * E_EDGES) return;
    int r = i / E_EDGES;
    atomicAdd(&s_out[r * N_NODES + src[i]], 1.0f);
    atomicAdd(&s_in [r * N_NODES + dst[i]], 1.0f);
}

// deg -> rsqrt(max(deg, 1))
__global__ void rgcn_degree_finalize(float* __restrict__ s, long n) {
    long i = (long)blockIdx.x * blockDim.x + threadIdx.x;
    if (i >= n) return;
    s[i] = rsqrtf(fmaxf(s[i], 1.0f));
}

// ---------------- edge scatter: agg[dst] += h[src] * s_out[src] ----------------
// one wave (32 lanes) per edge; each lane moves a float4 (128 floats / edge)
__global__ void rgcn_scatter(const float* __restrict__ h,
                             const int* __restrict__ src,
                             const int* __restrict__ dst,
                             const float* __restrict__ s_out,
                             float* __restrict__ agg) {
    int gid = blockIdx.x * blockDim.x + threadIdx.x;
    int e  = gid >> 5;
    if (e >= E_EDGES) return;
    int c4 = (gid & 31) << 2;
    int s = src[e];
    int d = dst[e];
    float sc = s_out[s];
    const float4 v = *(const float4*)(h + (long)s * DIM + c4);
    float* ap = agg + (long)d * DIM + c4;
    atomicAdd(ap + 0, v.x * sc);
    atomicAdd(ap + 1, v.y * sc);
    atomicAdd(ap + 2, v.z * sc);
    atomicAdd(ap + 3, v.w * sc);
}

// ---------------- per-relation GEMM with fp32 WMMA ----------------
// out[row0:row0+16, :] (+)= (agg[rows,:] * s_in[rows]) @ W  (+ bias)
// block: 256 threads = 8 waves; wave w owns output columns [16w, 16w+16)
// A tile staged in LDS, stride padded to 132 dwords (bank-conflict free).
__global__ void __launch_bounds__(256) rgcn_rel_gemm(
        const float* __restrict__ agg,
        const float* __restrict__ s_in,
        const float* __restrict__ W,     // [128,128] row-major: W[k*128 + n]
        const float* __restrict__ bias,  // [128]
        float* __restrict__ out,         // [N,128]
        int first, int do_relu) {
    __shared__ float tile[16 * 132];

    const int row0 = blockIdx.x * 16;
    const int t = threadIdx.x;

    // stage scaled A tile (16 x 128), coalesced; each thread fills 8 floats
#pragma unroll
    for (int i = 0; i < 8; ++i) {
        int idx = t + i * 256;
        int rr = idx >> 7;        // 0..15
        int cc = idx & 127;       // 0..127
        int row = row0 + rr;
        tile[rr * 132 + cc] = agg[(long)row * DIM + cc] * s_in[row];
    }
    __syncthreads();

    const int wave = t >> 5;
    const int lane = t & 31;
    const int m    = lane & 15;   // A row (M) / B column (N)
    const int hi   = lane >> 4;   // 0: K=0,1 ; 1: K=2,3
    const int col  = wave * 16 + m;

    // A fragment base: lane holds A[m][kb + 2*hi], A[m][kb + 2*hi + 1]
    const float* tp = tile + m * 132 + 2 * hi;
    // B fragment: lane holds W[kb + 2*hi][col], W[kb + 2*hi + 1][col]
    const float* wp = W + col;

    v8f c = {};
#pragma unroll
    for (int kb = 0; kb < DIM; kb += 4) {
        v2f a = *(const v2f*)(tp + kb);           // ds_load_b64, 8B aligned
        v2f b;
        b.x = wp[(kb + 2 * hi)     * DIM];
        b.y = wp[(kb + 2 * hi + 1) * DIM];
        c = __builtin_amdgcn_wmma_f32_16x16x4_f32(
                /*neg_a=*/false, a, /*neg_b=*/false, b,
                /*c_mod=*/(short)0, c, /*reuse_a=*/false, /*reuse_b=*/false);
    }

    // epilogue: C/D layout -> VGPR j holds row (j + 8*hi), lane column = col
    const float bsum = bias[col];
#pragma unroll
    for (int j = 0; j < 8; ++j) {
        int row = row0 + j + 8 * hi;
        long o = (long)row * DIM + col;
        float v = c[j] + bsum + (first ? 0.0f : out[o]);
        if (do_relu) v = fmaxf(v, 0.0f);
        out[o] = v;
    }
}

// ---------------- host launcher ----------------
extern "C" void kernel_launch(void* const* d_in, const int* in_sizes, int n_in,
                              void* d_out, int out_size, void* d_ws, size_t ws_size,
                              hipStream_t stream) {
    const float* x        = (const float*)d_in[0];           // [N,128]
    const int*   edge_src = (const int*)  d_in[1];           // [R,E]
    const int*   edge_dst = (const int*)  d_in[2];           // [R,E]
    const float* W1       = (const float*)d_in[3];           // [R,128,128]
    const float* b1       = (const float*)d_in[4];           // [R,128]
    const float* W2       = (const float*)d_in[5];           // [R,128,128]
    const float* b2       = (const float*)d_in[6];           // [R,128]
    float* out = (float*)d_out;                              // [N,128]

    // workspace layout (floats)
    float* s_out = (float*)d_ws;                             // R*N
    float* s_in  = s_out + (long)R_REL * N_NODES;            // R*N
    float* agg   = s_in  + (long)R_REL * N_NODES;            // N*128 (reused)
    float* h1    = agg   + (long)N_NODES * DIM;              // N*128

    // ---- degrees (graph is identical for both layers) ----
    {
        long nS = 2L * R_REL * N_NODES;
        rgcn_zero_f32<<<1024, 256, 0, stream>>>(s_out, nS);
        int ne = R_REL * E_EDGES;
        rgcn_degree_count<<<(ne + 255) / 256, 256, 0, stream>>>(edge_src, edge_dst, s_out, s_in);
        rgcn_degree_finalize<<<(int)((nS + 255) / 256), 256, 0, stream>>>(s_out, nS);
    }

    const long aggN = (long)N_NODES * DIM;
    const int scatter_blocks = (E_EDGES * 32 + 255) / 256;
    const int gemm_blocks = N_NODES / 16;   // 3125, exact

    for (int layer = 0; layer < 2; ++layer) {
        const float* hin  = layer ? (const float*)h1 : x;
        float*       hout = layer ? out : h1;
        const float* W    = layer ? W2 : W1;
        const float* bias = layer ? b2 : b1;

        for (int r = 0; r < R_REL; ++r) {
            rgcn_zero_f32<<<2048, 256, 0, stream>>>(agg, aggN);
            rgcn_scatter<<<scatter_blocks, 256, 0, stream>>>(
                hin, edge_src + (long)r * E_EDGES, edge_dst + (long)r * E_EDGES,
                s_out + (long)r * N_NODES, agg);
            rgcn_rel_gemm<<<gemm_blocks, 256, 0, stream>>>(
                agg, s_in + (long)r * N_NODES,
                W + (long)r * DIM * DIM, bias + (long)r * DIM,
                hout, /*first=*/(r == 0) ? 1 : 0,
                /*do_relu=*/(layer == 0 && r == R_REL - 1) ? 1 : 0);
        }
    }
}